// HAN_77567109365977
// MI455X (gfx1250) — compile-verified
//
#include <hip/hip_runtime.h>

#define HEADS 8
#define DIMH 16
#define HID 128
#define IN_F 256
#define NEG_SLOPE 0.2f

typedef __attribute__((ext_vector_type(2))) float v2f;
typedef __attribute__((ext_vector_type(8))) float v8f;

// ---- monotone float<->int mapping for atomic max on floats ----
__device__ __forceinline__ int han_fmap(float x) {
  int b = __float_as_int(x);
  return b >= 0 ? b : (b ^ 0x7fffffff);
}
__device__ __forceinline__ float han_funmap(int m) {
  int b = m >= 0 ? m : (m ^ 0x7fffffff);
  return __int_as_float(b);
}
__device__ __forceinline__ float han_lrelu(float v) {
  return v > 0.f ? v : NEG_SLOPE * v;
}

// =====================================================================
// Kernel: fill int buffer (emax init to INT_MIN)
// =====================================================================
__global__ void han_fill_int(int* __restrict__ p, int v, int n) {
  int i = blockIdx.x * blockDim.x + threadIdx.x;
  if (i < n) p[i] = v;
}

// =====================================================================
// Kernel: projection GEMM  xp = x @ Wp + bp   [N,256]@[256,128]
// One wave computes a 16x128 row tile with 8 fp32 WMMA accumulators.
// =====================================================================
__global__ void han_proj_wmma(const float* __restrict__ x,
                              const float* __restrict__ Wp,
                              const float* __restrict__ bp,
                              float* __restrict__ xp, int nTiles) {
  const int wave = (blockIdx.x * blockDim.x + threadIdx.x) >> 5;
  if (wave >= nTiles) return;                 // whole-wave exit: EXEC stays full
  const int lane = threadIdx.x & 31;
  const int half = lane >> 4;                 // 0: K lo pair, 1: K hi pair
  const int mn   = lane & 15;                 // A row / B-D column
  const int rowbase = wave * 16;

  v8f acc[8];
  #pragma unroll
  for (int t = 0; t < 8; ++t) {
    #pragma unroll
    for (int r = 0; r < 8; ++r) acc[t][r] = 0.0f;
  }

  const float* arow = x + (size_t)(rowbase + mn) * IN_F;
  for (int k = 0; k < IN_F; k += 4) {
    v2f a;
    const float* ap = arow + k + half * 2;
    a[0] = ap[0];
    a[1] = ap[1];
    #pragma unroll
    for (int t = 0; t < 8; ++t) {
      v2f b;
      b[0] = Wp[(size_t)(k + half * 2) * HID + t * 16 + mn];
      b[1] = Wp[(size_t)(k + half * 2 + 1) * HID + t * 16 + mn];
      acc[t] = __builtin_amdgcn_wmma_f32_16x16x4_f32(
          false, a, false, b, (short)0, acc[t], false, false);
    }
  }

  #pragma unroll
  for (int t = 0; t < 8; ++t) {
    float bias = bp[t * 16 + mn];
    #pragma unroll
    for (int r = 0; r < 8; ++r) {
      int row = rowbase + half * 8 + r;
      xp[(size_t)row * HID + t * 16 + mn] = acc[t][r] + bias;
    }
  }
}

// =====================================================================
// Kernel: per-(node,head) attention logits: 4 dot products of length 16
// =====================================================================
__global__ void han_alpha(const float* __restrict__ xp,
                          const float* __restrict__ as0, const float* __restrict__ ad0,
                          const float* __restrict__ as1, const float* __restrict__ ad1,
                          float* __restrict__ aS0, float* __restrict__ aD0,
                          float* __restrict__ aS1, float* __restrict__ aD1, int NH) {
  __shared__ float att[4][HID];
  for (int i = threadIdx.x; i < HID; i += blockDim.x) {
    att[0][i] = as0[i]; att[1][i] = ad0[i];
    att[2][i] = as1[i]; att[3][i] = ad1[i];
  }
  __syncthreads();
  int id = blockIdx.x * blockDim.x + threadIdx.x;
  if (id >= NH) return;
  int n = id >> 3, h = id & 7;
  const float* v = xp + (size_t)n * HID + h * DIMH;
  float s0 = 0.f, s1 = 0.f, s2 = 0.f, s3 = 0.f;
  #pragma unroll
  for (int d = 0; d < DIMH; ++d) {
    float xv = v[d];
    s0 += xv * att[0][h * DIMH + d];
    s1 += xv * att[1][h * DIMH + d];
    s2 += xv * att[2][h * DIMH + d];
    s3 += xv * att[3][h * DIMH + d];
  }
  aS0[id] = s0; aD0[id] = s1; aS1[id] = s2; aD1[id] = s3;
}

// =====================================================================
// Kernel: edge pass 1 — segment max via int-mapped atomicMax
// =====================================================================
__global__ void han_edge_max(const int* __restrict__ ei,
                             const float* __restrict__ aS, const float* __restrict__ aD,
                             int* __restrict__ emax, int E) {
  int e = blockIdx.x * blockDim.x + threadIdx.x;
  if (e >= E) return;
  int s = ei[e], d = ei[E + e];
  #pragma unroll
  for (int h = 0; h < HEADS; ++h) {
    float v = han_lrelu(aS[(size_t)s * HEADS + h] + aD[(size_t)d * HEADS + h]);
    atomicMax(&emax[(size_t)d * HEADS + h], han_fmap(v));
  }
}

// =====================================================================
// Kernel: edge pass 2 — denom[dst,h] += exp(e - emax)
// =====================================================================
__global__ void han_edge_sum(const int* __restrict__ ei,
                             const float* __restrict__ aS, const float* __restrict__ aD,
                             const int* __restrict__ emax, float* __restrict__ den, int E) {
  int e = blockIdx.x * blockDim.x + threadIdx.x;
  if (e >= E) return;
  int s = ei[e], d = ei[E + e];
  #pragma unroll
  for (int h = 0; h < HEADS; ++h) {
    float v = han_lrelu(aS[(size_t)s * HEADS + h] + aD[(size_t)d * HEADS + h]);
    float w = expf(v - han_funmap(emax[(size_t)d * HEADS + h]));
    atomicAdd(&den[(size_t)d * HEADS + h], w);
  }
}

// =====================================================================
// Kernel: edge pass 3 — wave per edge: out[dst] += alpha * xp[src]
// lane covers 4 contiguous floats (one head per 4 lanes)
// =====================================================================
__global__ void han_edge_aggr(const int* __restrict__ ei,
                              const float* __restrict__ aS, const float* __restrict__ aD,
                              const int* __restrict__ emax, const float* __restrict__ den,
                              const float* __restrict__ xp, float* __restrict__ outm, int E) {
  int wid = (blockIdx.x * blockDim.x + threadIdx.x) >> 5;
  int lane = threadIdx.x & 31;
  if (wid >= E) return;
  int s = ei[wid], d = ei[E + wid];
  int h = lane >> 2;
  float v = han_lrelu(aS[(size_t)s * HEADS + h] + aD[(size_t)d * HEADS + h]);
  float w = expf(v - han_funmap(emax[(size_t)d * HEADS + h]));
  float alpha = w / (den[(size_t)d * HEADS + h] + 1e-16f);
  const float4* src4 = (const float4*)(xp + (size_t)s * HID);
  float4 xv = src4[lane];
  float* o = outm + (size_t)d * HID + lane * 4;
  atomicAdd(o + 0, alpha * xv.x);
  atomicAdd(o + 1, alpha * xv.y);
  atomicAdd(o + 2, alpha * xv.z);
  atomicAdd(o + 3, alpha * xv.w);
}

// =====================================================================
// Kernel: semantic GEMM  tanh(relu(out_m) @ Wk + bk), column-sum into ksum
// Same WMMA tiling as the projection (K=128).
// =====================================================================
__global__ void han_semantic_wmma(const float* __restrict__ outm,
                                  const float* __restrict__ Wk,
                                  const float* __restrict__ bk,
                                  float* __restrict__ ksum, int nTiles) {
  const int wave = (blockIdx.x * blockDim.x + threadIdx.x) >> 5;
  if (wave >= nTiles) return;
  const int lane = threadIdx.x & 31;
  const int half = lane >> 4;
  const int mn   = lane & 15;
  const int rowbase = wave * 16;

  v8f acc[8];
  #pragma unroll
  for (int t = 0; t < 8; ++t) {
    #pragma unroll
    for (int r = 0; r < 8; ++r) acc[t][r] = 0.0f;
  }

  const float* arow = outm + (size_t)(rowbase + mn) * HID;
  for (int k = 0; k < HID; k += 4) {
    v2f a;
    a[0] = fmaxf(arow[k + half * 2], 0.f);      // fold ReLU into A load
    a[1] = fmaxf(arow[k + half * 2 + 1], 0.f);
    #pragma unroll
    for (int t = 0; t < 8; ++t) {
      v2f b;
      b[0] = Wk[(size_t)(k + half * 2) * HID + t * 16 + mn];
      b[1] = Wk[(size_t)(k + half * 2 + 1) * HID + t * 16 + mn];
      acc[t] = __builtin_amdgcn_wmma_f32_16x16x4_f32(
          false, a, false, b, (short)0, acc[t], false, false);
    }
  }

  #pragma unroll
  for (int t = 0; t < 8; ++t) {
    float bias = bk[t * 16 + mn];
    float s = 0.f;
    #pragma unroll
    for (int r = 0; r < 8; ++r) s += tanhf(acc[t][r] + bias);
    atomicAdd(&ksum[t * 16 + mn], s);
  }
}

// =====================================================================
// Kernel: semantic attention softmax over M=2 metapaths (tiny, serial)
// =====================================================================
__global__ void han_attn(const float* __restrict__ ksum, const float* __restrict__ q,
                         float* __restrict__ attn, float invN) {
  if (threadIdx.x == 0 && blockIdx.x == 0) {
    float d0 = 0.f, d1 = 0.f;
    for (int j = 0; j < HID; ++j) {
      d0 += ksum[j] * invN * q[j];
      d1 += ksum[HID + j] * invN * q[j];
    }
    float mx = fmaxf(d0, d1);
    float e0 = expf(d0 - mx), e1 = expf(d1 - mx);
    float s = e0 + e1;
    attn[0] = e0 / s;
    attn[1] = e1 / s;
  }
}

// =====================================================================
// Kernel: fused output  y = (a0*relu(out0)+a1*relu(out1)) @ lin_w + lin_b
// =====================================================================
__global__ void han_final(const float* __restrict__ out0, const float* __restrict__ out1,
                          const float* __restrict__ attn,
                          const float* __restrict__ lw, const float* __restrict__ lb,
                          float* __restrict__ y, int N) {
  __shared__ float W[HID * 4];
  __shared__ float A[2];
  for (int i = threadIdx.x; i < HID * 4; i += blockDim.x) W[i] = lw[i];
  if (threadIdx.x < 2) A[threadIdx.x] = attn[threadIdx.x];
  __syncthreads();
  int n = blockIdx.x * blockDim.x + threadIdx.x;
  if (n >= N) return;
  float a0 = A[0], a1 = A[1];
  float acc0 = lb[0], acc1 = lb[1], acc2 = lb[2], acc3 = lb[3];
  const float4* o0 = (const float4*)(out0 + (size_t)n * HID);
  const float4* o1 = (const float4*)(out1 + (size_t)n * HID);
  for (int jj = 0; jj < HID / 4; ++jj) {
    float4 u = o0[jj], v = o1[jj];
    float f[4];
    f[0] = a0 * fmaxf(u.x, 0.f) + a1 * fmaxf(v.x, 0.f);
    f[1] = a0 * fmaxf(u.y, 0.f) + a1 * fmaxf(v.y, 0.f);
    f[2] = a0 * fmaxf(u.z, 0.f) + a1 * fmaxf(v.z, 0.f);
    f[3] = a0 * fmaxf(u.w, 0.f) + a1 * fmaxf(v.w, 0.f);
    #pragma unroll
    for (int u4 = 0; u4 < 4; ++u4) {
      int j = jj * 4 + u4;
      acc0 += f[u4] * W[j * 4 + 0];
      acc1 += f[u4] * W[j * 4 + 1];
      acc2 += f[u4] * W[j * 4 + 2];
      acc3 += f[u4] * W[j * 4 + 3];
    }
  }
  float* yp = y + (size_t)n * 4;
  yp[0] = acc0; yp[1] = acc1; yp[2] = acc2; yp[3] = acc3;
}

// =====================================================================
extern "C" void kernel_launch(void* const* d_in, const int* in_sizes, int n_in,
                              void* d_out, int out_size, void* d_ws, size_t ws_size,
                              hipStream_t stream) {
  const float* x   = (const float*)d_in[0];
  const int*   ei0 = (const int*)d_in[1];
  const int*   ei1 = (const int*)d_in[2];
  const float* pw  = (const float*)d_in[3];
  const float* pb  = (const float*)d_in[4];
  const float* as0 = (const float*)d_in[5];
  const float* ad0 = (const float*)d_in[6];
  const float* as1 = (const float*)d_in[7];
  const float* ad1 = (const float*)d_in[8];
  const float* klw = (const float*)d_in[9];
  const float* klb = (const float*)d_in[10];
  const float* q   = (const float*)d_in[11];
  const float* lw  = (const float*)d_in[12];
  const float* lb  = (const float*)d_in[13];
  float* y = (float*)d_out;

  const int N  = in_sizes[0] / IN_F;     // 100000
  const int E0 = in_sizes[1] / 2;        // 1600000
  const int E1 = in_sizes[2] / 2;
  const size_t NH = (size_t)N * HEADS;

  // ---- workspace layout (floats) ----
  float* xp   = (float*)d_ws;            // N*128
  float* aS0  = xp + (size_t)N * HID;    // N*8 each
  float* aD0  = aS0 + NH;
  float* aS1  = aD0 + NH;
  float* aD1  = aS1 + NH;
  int*   emx  = (int*)(aD1 + NH);        // 2 * N*8 (int-mapped max)
  float* den  = (float*)(emx + 2 * NH);  // 2 * N*8
  float* out0 = den + 2 * NH;            // N*128
  float* out1 = out0 + (size_t)N * HID;  // N*128
  float* ksum = out1 + (size_t)N * HID;  // 2*128
  float* attn = ksum + 2 * HID;          // 2

  // zero den..attn (contiguous), init emax to INT_MIN
  size_t zf = 2 * NH + 2 * (size_t)N * HID + 2 * HID + 2;
  hipMemsetAsync(den, 0, zf * sizeof(float), stream);
  int nem = (int)(2 * NH);
  han_fill_int<<<(nem + 255) / 256, 256, 0, stream>>>(emx, (int)0x80000000, nem);

  const int nTiles = N / 16;             // N = 100000 -> 6250 exact
  const int gemmBlocks = (nTiles * 32 + 255) / 256;

  han_proj_wmma<<<gemmBlocks, 256, 0, stream>>>(x, pw, pb, xp, nTiles);
  han_alpha<<<((int)NH + 255) / 256, 256, 0, stream>>>(xp, as0, ad0, as1, ad1,
                                                       aS0, aD0, aS1, aD1, (int)NH);

  // metapath 0
  han_edge_max<<<(E0 + 255) / 256, 256, 0, stream>>>(ei0, aS0, aD0, emx, E0);
  han_edge_sum<<<(E0 + 255) / 256, 256, 0, stream>>>(ei0, aS0, aD0, emx, den, E0);
  han_edge_aggr<<<(unsigned)(((size_t)E0 * 32 + 255) / 256), 256, 0, stream>>>(
      ei0, aS0, aD0, emx, den, xp, out0, E0);

  // metapath 1
  int* emx1 = emx + NH;
  float* den1 = den + NH;
  han_edge_max<<<(E1 + 255) / 256, 256, 0, stream>>>(ei1, aS1, aD1, emx1, E1);
  han_edge_sum<<<(E1 + 255) / 256, 256, 0, stream>>>(ei1, aS1, aD1, emx1, den1, E1);
  han_edge_aggr<<<(unsigned)(((size_t)E1 * 32 + 255) / 256), 256, 0, stream>>>(
      ei1, aS1, aD1, emx1, den1, xp, out1, E1);

  // semantic attention
  han_semantic_wmma<<<gemmBlocks, 256, 0, stream>>>(out0, klw, klb, ksum, nTiles);
  han_semantic_wmma<<<gemmBlocks, 256, 0, stream>>>(out1, klw, klb, ksum + HID, nTiles);
  han_attn<<<1, 32, 0, stream>>>(ksum, q, attn, 1.0f / (float)N);

  // fused output projection
  han_final<<<(N + 255) / 256, 256, 0, stream>>>(out0, out1, attn, lw, lb, y, N);
}